// TwoCompColumn_38869454029213
// MI455X (gfx1250) — compile-verified
//
#include <hip/hip_runtime.h>
#include <cstdint>

// ---------------------------------------------------------------------------
// Problem constants (match reference)
// ---------------------------------------------------------------------------
static constexpr int N_CH      = 2000000;
static constexpr int N_PER     = 6;
static constexpr int N_TOTAL   = N_CH * N_PER;   // 12,000,000
static constexpr int N_HALF    = N_TOTAL / 2;    // 6,000,000 : threefry counter split
static constexpr int N_THREADS = N_CH / 2;       // 1,000,000 : each thread owns 2 channels
static constexpr int BLOCK     = 256;

// ---------------------------------------------------------------------------
// threefry2x32 (JAX's default PRNG), constexpr so the 10 substep keys from
// jax.random.split(jax.random.key(42), 10) are computed at compile time.
// ---------------------------------------------------------------------------
struct TF2 { unsigned a, b; };

#define TF_ROUND(r) { x0 += x1; x1 = ((x1 << (r)) | (x1 >> (32 - (r)))) ^ x0; }

__host__ __device__ constexpr TF2 threefry2x32(unsigned k0, unsigned k1,
                                               unsigned c0, unsigned c1) {
  unsigned ks2 = 0x1BD11BDAu ^ k0 ^ k1;
  unsigned x0 = c0 + k0;
  unsigned x1 = c1 + k1;
  TF_ROUND(13) TF_ROUND(15) TF_ROUND(26) TF_ROUND(6)
  x0 += k1;  x1 += ks2 + 1u;
  TF_ROUND(17) TF_ROUND(29) TF_ROUND(16) TF_ROUND(24)
  x0 += ks2; x1 += k0 + 2u;
  TF_ROUND(13) TF_ROUND(15) TF_ROUND(26) TF_ROUND(6)
  x0 += k0;  x1 += k1 + 3u;
  TF_ROUND(17) TF_ROUND(29) TF_ROUND(16) TF_ROUND(24)
  x0 += k1;  x1 += ks2 + 4u;
  TF_ROUND(13) TF_ROUND(15) TF_ROUND(26) TF_ROUND(6)
  x0 += ks2; x1 += k0 + 5u;
  return TF2{x0, x1};
}
#undef TF_ROUND

struct Keys { unsigned k0[10]; unsigned k1[10]; };

__host__ __device__ constexpr Keys make_substep_keys() {
  // jax.random.key(42) -> (0, 42).  split(key,10): threefry(key, iota(20))
  // with x0=[0..9], x1=[10..19]; flat = concat(out0,out1); key[i]=(flat[2i],flat[2i+1]).
  unsigned o0[10] = {}, o1[10] = {};
  for (int i = 0; i < 10; ++i) {
    TF2 t = threefry2x32(0u, 42u, (unsigned)i, (unsigned)(10 + i));
    o0[i] = t.a; o1[i] = t.b;
  }
  Keys K{};
  for (int i = 0; i < 10; ++i) {
    int f0 = 2 * i, f1 = 2 * i + 1;
    K.k0[i] = (f0 < 10) ? o0[f0] : o1[f0 - 10];
    K.k1[i] = (f1 < 10) ? o0[f1] : o1[f1 - 10];
  }
  return K;
}

__device__ __constant__ Keys g_keys = make_substep_keys();

// ---------------------------------------------------------------------------
// JAX bit->uniform(-1,1) mapping and branchless single-precision erfinv
// (XLA's f32 erf_inv polynomial) so normal = sqrt(2)*erfinv(u) matches
// jax.random.normal closely.
// ---------------------------------------------------------------------------
__device__ __forceinline__ float bits_to_u(unsigned b) {
  float f = __uint_as_float((b >> 9) | 0x3f800000u) - 1.0f;  // [0,1)
  const float lo = -0.99999994f;                              // nextafter(-1,0)
  float u = fmaf(f, 1.99999994f, lo);                         // (hi-lo)*f + lo
  return fmaxf(u, lo);
}

__device__ __forceinline__ float erfinv_f(float u) {
  // XLA: w = -log1p(-u*u); approximate arg with one fused rounding.
  float one_m_u2 = fmaf(-u, u, 1.0f);
  float w  = -__logf(fmaxf(one_m_u2, 1e-38f));
  float wc = w - 2.5f;
  float ws = sqrtf(w) - 3.0f;
  float p1 = 2.81022636e-08f;
  p1 = fmaf(p1, wc, 3.43273939e-07f);
  p1 = fmaf(p1, wc, -3.5233877e-06f);
  p1 = fmaf(p1, wc, -4.39150654e-06f);
  p1 = fmaf(p1, wc, 0.00021858087f);
  p1 = fmaf(p1, wc, -0.00125372503f);
  p1 = fmaf(p1, wc, -0.00417768164f);
  p1 = fmaf(p1, wc, 0.246640727f);
  p1 = fmaf(p1, wc, 1.50140941f);
  float p2 = -0.000200214257f;
  p2 = fmaf(p2, ws, 0.000100950558f);
  p2 = fmaf(p2, ws, 0.00134934322f);
  p2 = fmaf(p2, ws, -0.00367342844f);
  p2 = fmaf(p2, ws, 0.00573950773f);
  p2 = fmaf(p2, ws, -0.0076224613f);
  p2 = fmaf(p2, ws, 0.00943887047f);
  p2 = fmaf(p2, ws, 1.00167406f);
  p2 = fmaf(p2, ws, 2.83297682f);
  float p = (w < 5.0f) ? p1 : p2;   // branchless select, no divergence
  return p * u;
}

__device__ __forceinline__ float sigmoidf(float x) {
  return 1.0f / (1.0f + __expf(-x));
}

__device__ __forceinline__ void izh_substep(float& v, float& u, float& r, float I) {
  float vh = v + 0.5f * (0.04f * v * v + 5.0f * v + 140.0f - u + I);
  float uh = u + 0.01f * (0.2f * v - u);                 // 0.5*A, A=0.02, B=0.2
  float vn = vh + 0.5f * (0.04f * vh * vh + 5.0f * vh + 140.0f - uh + I);
  float un = uh + 0.01f * (0.2f * vh - uh);
  vn = fminf(fmaxf(vn, -100.0f), 35.0f);
  bool fired = (vn >= 30.0f);
  v = fired ? -65.0f : vn;                               // C
  u = fired ? (un + 8.0f) : un;                          // +D
  r = fmaf(r, 0.98f, fired ? 0.02f : 0.0f);              // TAU_RATE
}

// ---------------------------------------------------------------------------
// CDNA5 async global->LDS copy (ASYNCcnt path, gfx1250-only instruction).
// GVS mode: addr = SGPR64 base + VGPR32 byte-offset.
// ---------------------------------------------------------------------------
__device__ __forceinline__ void async_ld_f32(float* lds_slot, const float* gbase,
                                             unsigned elem) {
  unsigned loff = (unsigned)(unsigned long long)lds_slot;  // low 32b of generic = LDS offset
  unsigned goff = elem * 4u;
  asm volatile("global_load_async_to_lds_b32 %0, %1, %2"
               :: "v"(loff), "v"(goff), "s"(gbase) : "memory");
}

__device__ __forceinline__ void async_wait0() {
  asm volatile("s_wait_asynccnt 0" ::: "memory");
}

// ---------------------------------------------------------------------------
// Kernel: one thread handles channels (t, t+1M) so its 6 neuron pairs
// (i, i+6M) are exactly JAX's threefry counter pairs -> 1 threefry = 2 noises.
// ---------------------------------------------------------------------------
__global__ __launch_bounds__(BLOCK) void TwoCompColumn_kernel(
    const float* __restrict__ sens, const float* __restrict__ pred,
    const float* __restrict__ vs,   const float* __restrict__ us,
    const float* __restrict__ vas,  const float* __restrict__ rts,
    const float* __restrict__ gam,  const float* __restrict__ att,
    const float* __restrict__ bias, float* __restrict__ out) {
  const int tid = threadIdx.x;
  const int t = blockIdx.x * BLOCK + tid;
  if (t >= N_THREADS) return;

  const int c0 = t, c1 = t + N_THREADS;
  const int b0 = t * N_PER;            // < 6M
  const int b1 = b0 + N_HALF;          // >= 6M

  // ---- stage the 10 per-channel scalars via the async DMA path ----
  __shared__ float s_ch[10 * BLOCK];
  async_ld_f32(&s_ch[0 * BLOCK + tid], sens, (unsigned)c0);
  async_ld_f32(&s_ch[1 * BLOCK + tid], sens, (unsigned)c1);
  async_ld_f32(&s_ch[2 * BLOCK + tid], pred, (unsigned)c0);
  async_ld_f32(&s_ch[3 * BLOCK + tid], pred, (unsigned)c1);
  async_ld_f32(&s_ch[4 * BLOCK + tid], gam,  (unsigned)c0);
  async_ld_f32(&s_ch[5 * BLOCK + tid], gam,  (unsigned)c1);
  async_ld_f32(&s_ch[6 * BLOCK + tid], att,  (unsigned)c0);
  async_ld_f32(&s_ch[7 * BLOCK + tid], att,  (unsigned)c1);
  async_ld_f32(&s_ch[8 * BLOCK + tid], bias, (unsigned)c0);
  async_ld_f32(&s_ch[9 * BLOCK + tid], bias, (unsigned)c1);

  // ---- per-neuron state into registers (coalesced, overlapped w/ async) ----
  float v[12], u[12], a[12], r[12], I[12];
#pragma unroll
  for (int j = 0; j < N_PER; ++j) {
    v[j] = vs[b0 + j];  v[6 + j] = vs[b1 + j];
    u[j] = us[b0 + j];  u[6 + j] = us[b1 + j];
    a[j] = vas[b0 + j]; a[6 + j] = vas[b1 + j];
    r[j] = rts[b0 + j]; r[6 + j] = rts[b1 + j];
  }

  async_wait0();  // each thread reads only its own staged slots: no barrier needed
  const float se0 = s_ch[0 * BLOCK + tid], se1 = s_ch[1 * BLOCK + tid];
  const float pr0 = s_ch[2 * BLOCK + tid], pr1 = s_ch[3 * BLOCK + tid];
  const float ga0 = s_ch[4 * BLOCK + tid], ga1 = s_ch[5 * BLOCK + tid];
  const float at0 = s_ch[6 * BLOCK + tid], at1 = s_ch[7 * BLOCK + tid];
  const float bb0 = s_ch[8 * BLOCK + tid], bb1 = s_ch[9 * BLOCK + tid];

  const float pi0 = sigmoidf(ga0), pi1 = sigmoidf(ga1);
  const float eb0 = bb0 + 2.0f * pi0 + 0.1f * at0;
  const float eb1 = bb1 + 2.0f * pi1 + 0.1f * at1;
  const float Ic0 = fmaf(se0, 12.0f, eb0);
  const float Ic1 = fmaf(se1, 12.0f, eb1);

  // apical integration + per-neuron input current
#pragma unroll
  for (int j = 0; j < 12; ++j) {
    float pd = (j < 6) ? pr0 : pr1;
    float aa = fminf(fmaxf(fmaf(0.65f, a[j], 0.35f * pd), -2.0f), 2.0f);
    a[j] = aa;
    I[j] = ((j < 6) ? Ic0 : Ic1) + 7.5f * (sigmoidf(aa) - 0.5f);  // 0.5*(sig-0.5)*15
  }

  // ---- 10 substeps; one threefry per neuron-pair ----
  const float NSCALE = 0.3f * 1.41421356f;  // 0.3 * sqrt(2)
  for (int s = 0; s < 10; ++s) {
    const unsigned k0 = g_keys.k0[s];
    const unsigned k1 = g_keys.k1[s];
#pragma unroll
    for (int j = 0; j < N_PER; ++j) {
      TF2 bits = threefry2x32(k0, k1, (unsigned)(b0 + j), (unsigned)(b1 + j));
      float n0 = NSCALE * erfinv_f(bits_to_u(bits.a));
      float n1 = NSCALE * erfinv_f(bits_to_u(bits.b));
      izh_substep(v[j],     u[j],     r[j],     I[j] + n0);
      izh_substep(v[6 + j], u[6 + j], r[6 + j], I[6 + j] + n1);
    }
  }

  // ---- per-channel pooling (pure register reduction) ----
  float sva0 = 0.f, svn0 = 0.f, sr0 = 0.f;
  float sva1 = 0.f, svn1 = 0.f, sr1 = 0.f;
#pragma unroll
  for (int j = 0; j < N_PER; ++j) {
    float vn0 = fminf(fmaxf((v[j] + 65.0f) * (1.0f / 95.0f), 0.0f), 1.0f);
    float vn1 = fminf(fmaxf((v[6 + j] + 65.0f) * (1.0f / 95.0f), 0.0f), 1.0f);
    sva0 += a[j];     svn0 += vn0; sr0 += r[j];
    sva1 += a[6 + j]; svn1 += vn1; sr1 += r[6 + j];
  }
  const float inv6 = 1.0f / 6.0f;
  const float pe0 = (sva0 - svn0) * inv6;
  const float pe1 = (sva1 - svn1) * inv6;

  float sc0 = 0.f, sc1 = 0.f;
#pragma unroll
  for (int j = 0; j < N_PER; ++j) {
    sc0 += fminf(fmaxf(a[j]     - 0.05f * pe0, -2.0f), 2.0f);
    sc1 += fminf(fmaxf(a[6 + j] - 0.05f * pe1, -2.0f), 2.0f);
  }

  // ---- outputs: (5, N_CH) row-major ----
  out[0 * N_CH + c0] = pe0;        out[0 * N_CH + c1] = pe1;
  out[1 * N_CH + c0] = pi0;        out[1 * N_CH + c1] = pi1;
  out[2 * N_CH + c0] = sr0 * inv6; out[2 * N_CH + c1] = sr1 * inv6;
  out[3 * N_CH + c0] = sc0 * inv6; out[3 * N_CH + c1] = sc1 * inv6;
  out[4 * N_CH + c0] = eb0;        out[4 * N_CH + c1] = eb1;
}

// ---------------------------------------------------------------------------
extern "C" void kernel_launch(void* const* d_in, const int* in_sizes, int n_in,
                              void* d_out, int out_size, void* d_ws, size_t ws_size,
                              hipStream_t stream) {
  const float* sens = (const float*)d_in[0];
  const float* pred = (const float*)d_in[1];
  const float* vs   = (const float*)d_in[2];
  const float* us   = (const float*)d_in[3];
  const float* vas  = (const float*)d_in[4];
  const float* rts  = (const float*)d_in[5];
  const float* gam  = (const float*)d_in[6];
  const float* att  = (const float*)d_in[7];
  const float* bias = (const float*)d_in[8];
  // d_in[9] (prev_pe_mag) is unused by the reference computation.
  float* out = (float*)d_out;

  dim3 grid((N_THREADS + BLOCK - 1) / BLOCK), block(BLOCK);
  hipLaunchKernelGGL(TwoCompColumn_kernel, grid, block, 0, stream,
                     sens, pred, vs, us, vas, rts, gam, att, bias, out);
}